// loss_generator_26182120637003
// MI455X (gfx1250) — compile-verified
//
#include <hip/hip_runtime.h>
#include <cstdint>

// Wave-equation time stepping, persistent-kernel formulation for MI455X (gfx1250).
//
//   u0 = zero_boundary(input[0]);  u1 = u0
//   for t in 0..255:  out[t] = pad(2*uc - up + r*(5pt laplacian of uc), 1)
//
// One kernel, 48 WGs x 256 threads, atomic grid barrier between steps.
// LDS double buffer holds each WG's band; per step only the 2 halo rows are
// fetched from global memory, via the CDNA5 Tensor Data Mover (TENSORcnt).
// The one-time t==0 band fill uses per-lane async global->LDS copies (ASYNCcnt).

#define N        384
#define N2       (N * N)
#define TSTEPS   256
#define NWG      48
#define BROWS    8              // rows per workgroup band (48*8 = 384)
#define TROWS    (BROWS + 2)    // band + 2 halo rows
#define TILE_ELEMS (TROWS * N)  // 3840 floats per buffer
#define BLOCK    256

typedef unsigned int u32x4 __attribute__((ext_vector_type(4)));
typedef int          i32x8 __attribute__((ext_vector_type(8)));
typedef int          i32x4 __attribute__((ext_vector_type(4)));

__global__ void wave_init_counter(unsigned* counter) { *counter = 0u; }

// Issue one TDM descriptor: DMA a contiguous row of 384 f32 from global into LDS.
// D# layout per CDNA5 ISA ch.8 (group0: count/lds_addr/global_addr/type=2;
// group1: data_size=4B, tensor_dim0=384, tensor_dim1=1, tile_dim0=384, tile_dim1=1).
// This toolchain exposes the 6-arg builtin: (g0, g1, g2, g3, extra, cpol).
__device__ __forceinline__ void tdm_load_row(unsigned lds_byte, const float* gsrc) {
    const unsigned long long ga = (unsigned long long)(uintptr_t)gsrc;
    u32x4 g0;
    g0.x = 1u;                                        // count=1, user mode, no gather
    g0.y = lds_byte;                                  // lds_addr (bytes)
    g0.z = (unsigned)(ga & 0xFFFFFFFFull);            // global_addr[31:0]
    g0.w = (unsigned)((ga >> 32) & 0x1FFFFFFull)      // global_addr[56:32]
         | (2u << 30);                                // type=2 ("image")
    i32x8 g1;
    g1[0] = (int)(2u << 16);                          // wg_mask=0, data_size=2 (4B)
    g1[1] = (int)(384u << 16);                        // tensor_dim0[15:0]=384
    g1[2] = (int)(1u << 16);                          // tensor_dim0 hi=0, tensor_dim1=1
    g1[3] = (int)(384u << 16);                        // tensor_dim1 hi=0, tile_dim0=384
    g1[4] = 1;                                        // tile_dim1=1, tile_dim2=0
    g1[5] = 384;                                      // tensor_dim0_stride[31:0]
    g1[6] = 0;                                        // stride hi / dim1_stride lo
    g1[7] = 0;
    const i32x4 z4 = {0, 0, 0, 0};                    // groups 2/3: dims 3/4 unused
    const i32x8 z8 = {0, 0, 0, 0, 0, 0, 0, 0};
    __builtin_amdgcn_tensor_load_to_lds(g0, g1, z4, z4, z8, 0);
}

__global__ __launch_bounds__(BLOCK) void wave_persistent(
    const float* __restrict__ in0,    // input slice 0 (384*384)
    const float* __restrict__ cptr,   // c (1 element)
    float* __restrict__ out,          // 256*384*384
    unsigned* __restrict__ counter)   // grid-barrier counter (in d_ws)
{
    __shared__ float tile[2 * TILE_ELEMS];

    const int tid  = threadIdx.x;
    const int wg   = blockIdx.x;
    const int row0 = wg * BROWS;                  // first global row of band

    const float c = cptr[0];
    const float r = 1.0e-4f * c * c;              // DT^2*c^2/DX^2, DX=DY=1

    // Low 32 bits of the generic pointer to a __shared__ object are the
    // workgroup-relative LDS byte offset (aperture occupies the high bits).
    const unsigned lds_base = (unsigned)(unsigned long long)(uintptr_t)tile;

    for (int t = 0; t < TSTEPS; ++t) {
        const int cb = t & 1;
        float* cur = tile + (cb ? TILE_ELEMS : 0);  // uc band (+halo)
        float* nxt = tile + (cb ? 0 : TILE_ELEMS);  // holds out[t-2] band; becomes out[t]
        const unsigned cur_byte = lds_base + (cb ? TILE_ELEMS * 4u : 0u);

        // ---------------- stage uc into LDS ----------------
        if (t == 0) {
            // Full fill of u0 = zero_boundary(in0): interior elements via
            // per-lane async global->LDS copies, boundary/outside as zeros.
            for (int k = tid; k < TILE_ELEMS; k += BLOCK) {
                const int tr  = k / N;
                const int cix = k - tr * N;
                const int g   = row0 - 1 + tr;
                if (g > 0 && g < N - 1 && cix > 0 && cix < N - 1) {
                    const unsigned lo = cur_byte + (unsigned)k * 4u;
                    const unsigned long long gaddr =
                        (unsigned long long)(uintptr_t)(in0 + g * N + cix);
                    asm volatile("global_load_async_to_lds_b32 %0, %1, off"
                                 :: "v"(lo), "v"(gaddr)
                                 : "memory");
                } else {
                    cur[k] = 0.0f;
                }
            }
            asm volatile("s_wait_asynccnt 0" ::: "memory");
            __syncthreads();
        } else {
            // Rows 1..8 of `cur` were produced by step t-1's compute.
            // DMA the two halo rows from out[t-1] with the Tensor Data Mover.
            // Out-of-range halo rows are never read (adjacent rows are masked
            // boundary rows), so no zero-fill is required for them.
            const float* uc_g = out + (size_t)(t - 1) * N2;
            if (tid < 32) {                       // wave 0 only (uniform branch)
                if (row0 - 1 >= 0)
                    tdm_load_row(cur_byte, uc_g + (size_t)(row0 - 1) * N);
                if (row0 + BROWS <= N - 1)
                    tdm_load_row(cur_byte + (TROWS - 1) * N * 4u,
                                 uc_g + (size_t)(row0 + BROWS) * N);
                __builtin_amdgcn_s_wait_tensorcnt(0);
            }
            __syncthreads();
        }

        // ---------------- compute band, write out[t] ----------------
        for (int k = tid; k < BROWS * N; k += BLOCK) {
            const int rr_ = k / N;            // 0..7 within band
            const int cix = k - rr_ * N;
            const int g   = row0 + rr_;       // global row
            const int kk  = (rr_ + 1) * N + cix;

            float un = 0.0f;
            if (g > 0 && g < N - 1 && cix > 0 && cix < N - 1) {
                const float ui  = cur[kk];
                // up == uc at t==0; otherwise out[t-2] lives in `nxt`
                // (read-before-overwrite, each element touched by one thread).
                const float upv = (t == 0) ? ui : nxt[kk];
                const float lap = cur[kk + N] + cur[kk - N] +
                                  cur[kk + 1] + cur[kk - 1] - 4.0f * ui;
                un = 2.0f * ui - upv + r * lap;
            }
            nxt[kk] = un;                                   // feeds steps t+1, t+2
            out[(size_t)t * N2 + (size_t)g * N + cix] = un; // streamed to HBM
        }

        // ---------------- device-wide barrier ----------------
        __syncthreads();
        if (tid == 0) {
            __threadfence();  // publish this WG's out[t] stores at agent scope
            __hip_atomic_fetch_add(counter, 1u, __ATOMIC_RELEASE,
                                   __HIP_MEMORY_SCOPE_AGENT);
            const unsigned tgt = (unsigned)NWG * (unsigned)(t + 1);
            while (__hip_atomic_load(counter, __ATOMIC_ACQUIRE,
                                     __HIP_MEMORY_SCOPE_AGENT) < tgt) {
                __builtin_amdgcn_s_sleep(2);
            }
        }
        __syncthreads();
    }
}

extern "C" void kernel_launch(void* const* d_in, const int* in_sizes, int n_in,
                              void* d_out, int out_size, void* d_ws, size_t ws_size,
                              hipStream_t stream) {
    (void)in_sizes; (void)n_in; (void)out_size; (void)ws_size;
    const float* u_in = (const float*)d_in[0];   // (258,1,384,384); only slice 0 used
    const float* c    = (const float*)d_in[1];   // (1,1)
    float* out        = (float*)d_out;           // (256,1,384,384)
    unsigned* counter = (unsigned*)d_ws;

    wave_init_counter<<<1, 1, 0, stream>>>(counter);
    wave_persistent<<<NWG, BLOCK, 0, stream>>>(u_in, c, out, counter);
}